// DM_42417097016803
// MI455X (gfx1250) — compile-verified
//
#include <hip/hip_runtime.h>

// PV-DM scoring kernel for MI455X (gfx1250, wave32).
// scores[b,k] = (D[doc[b]] + sum_c W[ctx[b,c]]) . O[:, tgt[b,k]]
//
// Memory-bound gather (~60 MB -> ~2.6us at 23.3 TB/s HBM; O (51 MB) is
// L2-resident so the strided column gather is latency- not bandwidth-bound).
// Dot products run on the matrix pipe: per 16-row batch tile and fixed k,
// chain V_WMMA_F32_16X16X4_F32 (f32 end-to-end, matches reference precision)
// and extract the diagonal of the 16x16 f32 accumulator.
//
// Latency hiding: 4 waves/block -> wave (wk, wd): wk in {0,1} picks the
// k-triple, wd in {0,1} picks a 128-wide d-half (partials combined via LDS).
// 256 blocks x 4 waves = 1024 waves; unroll 8 keeps ~16 strided loads in
// flight per wave (LOADcnt allows 63; 24-bit offsets cover the 6.2 MB span).

#define VEC_DIM   256
#define NUM_WORDS 50000
#define BATCH     4096
#define NUM_CTX   8
#define NUM_K     6
#define TILE_B    16
#define THREADS   128

typedef __attribute__((ext_vector_type(2))) float v2f;
typedef __attribute__((ext_vector_type(8))) float v8f;

__global__ __launch_bounds__(THREADS) void pvdm_wmma_kernel(
    const int*   __restrict__ ctx,   // [BATCH][NUM_CTX]
    const int*   __restrict__ doc,   // [BATCH]
    const int*   __restrict__ tgt,   // [BATCH][NUM_K]
    const float* __restrict__ Dm,    // [NUM_DOCS][VEC_DIM]
    const float* __restrict__ Wm,    // [NUM_WORDS][VEC_DIM]
    const float* __restrict__ Om,    // [VEC_DIM][NUM_WORDS]
    float*       __restrict__ out)   // [BATCH][NUM_K]
{
    // x tile: 16 rows x 256 floats, rows padded to 260 floats (1040 B, 16B
    // aligned) so the phase-2 ds_load_2addr_b64 reads are bank-conflict-free.
    __shared__ __align__(16) float xs[TILE_B][VEC_DIM + 4];
    __shared__ float part[2][TILE_B];    // wd=1 partial diagonals, per wk

    const int b0   = blockIdx.x * TILE_B;
    const int tid  = threadIdx.x;         // 0..127
    const int lane = tid & 31;
    const int wave = tid >> 5;            // 0..3
    const int wk   = wave & 1;            // k-triple select
    const int wd   = wave >> 1;           // d-half select

    // ---------------- Phase 1: build x[16][256] into LDS ----------------
    // 16 rows x 64 float4-cols = 1024 tasks over 128 threads -> 8 iterations.
    for (int task = tid; task < TILE_B * (VEC_DIM / 4); task += THREADS) {
        const int row = task >> 6;        // 0..15
        const int c4  = task & 63;        // float4 column
        const int b   = b0 + row;
        const float4* dp = (const float4*)(Dm + (long)doc[b] * VEC_DIM) + c4;
        float4 acc = *dp;
        #pragma unroll
        for (int c = 0; c < NUM_CTX; ++c) {
            const float4* wp =
                (const float4*)(Wm + (long)ctx[b * NUM_CTX + c] * VEC_DIM) + c4;
            float4 wv = *wp;
            acc.x += wv.x; acc.y += wv.y; acc.z += wv.z; acc.w += wv.w;
        }
        *(float4*)&xs[row][c4 * 4] = acc;
    }
    __syncthreads();

    // ---------------- Phase 2: WMMA chains over this wave's d-half -------
    // 32-bit A (16x4) layout: lanes 0-15 hold {K=0,K=1}, lanes 16-31 {K=2,K=3}.
    // B (4x16) mirrored; C/D 16x16: vgpr r = row M=r (lanes 0-15) / M=r+8.
    const int row   = lane & 15;               // A row (M) / B col (N)
    const int kOff  = (lane < 16) ? 0 : 2;     // K-pair fed by this half-wave
    const int dBase = wd * (VEC_DIM / 2);      // 0 or 128

    for (int kk = 0; kk < 3; ++kk) {
        const int k   = wk * 3 + kk;
        const int col = tgt[(b0 + row) * NUM_K + k];     // O column for lane's N
        const float* ob = Om + (long)dBase * NUM_WORDS + col;

        v8f c = {0.f, 0.f, 0.f, 0.f, 0.f, 0.f, 0.f, 0.f};

        #pragma unroll 8
        for (int step = 0; step < VEC_DIM / 8; ++step) { // 32 steps x K=4 = 128 d
            const int d0 = step * 4 + kOff;
            v2f a, b;
            a.x = xs[row][dBase + d0];
            a.y = xs[row][dBase + d0 + 1];
            b.x = ob[(long)d0 * NUM_WORDS];
            b.y = ob[(long)(d0 + 1) * NUM_WORDS];
            // 8 args: (neg_a, A, neg_b, B, c_mod, C, reuse_a, reuse_b)
            c = __builtin_amdgcn_wmma_f32_16x16x4_f32(
                    false, a, false, b, (short)0, c, false, false);
        }

        // Partial diagonal D[i,i]:
        //   i in 0..7  -> lane i,    vgpr i
        //   i in 8..15 -> lane i+16, vgpr i-8
        float val = 0.f;
        int   di  = 0;
        bool  own = false;
        if (lane < 8)        { val = c[lane];      di = lane;      own = true; }
        else if (lane >= 24) { val = c[lane - 24]; di = lane - 16; own = true; }

        if (wd == 1 && own) part[wk][di] = val;
        __syncthreads();
        if (wd == 0 && own)
            out[(b0 + di) * NUM_K + k] = val + part[wk][di];
        __syncthreads();   // protect part[] before next iteration's write
    }
}

extern "C" void kernel_launch(void* const* d_in, const int* in_sizes, int n_in,
                              void* d_out, int out_size, void* d_ws, size_t ws_size,
                              hipStream_t stream) {
    (void)in_sizes; (void)n_in; (void)out_size; (void)d_ws; (void)ws_size;
    const int*   ctx = (const int*)  d_in[0];  // context_ids  [4096][8]
    const int*   doc = (const int*)  d_in[1];  // doc_ids      [4096]
    const int*   tgt = (const int*)  d_in[2];  // target_noise [4096][6]
    const float* Dm  = (const float*)d_in[3];  // D [100000][256]
    const float* Wm  = (const float*)d_in[4];  // W [50000][256]
    const float* Om  = (const float*)d_in[5];  // O [256][50000]
    float* out = (float*)d_out;                // [4096][6]

    dim3 grid(BATCH / TILE_B);   // 256 workgroups
    dim3 block(THREADS);         // 4 waves: (k-triple) x (d-half)
    hipLaunchKernelGGL(pvdm_wmma_kernel, grid, block, 0, stream,
                       ctx, doc, tgt, Dm, Wm, Om, out);
}